// DepthwiseMultiScaleFIR_31877247271515
// MI455X (gfx1250) — compile-verified
//
#include <hip/hip_runtime.h>
#include <hip/hip_bf16.h>
#include <stdint.h>

// ---------------- problem constants (match reference) ----------------
constexpr int Bn = 4, Ln = 2048, Hn = 16, Dn = 64;
constexpr int Cn = Hn * Dn;                 // 1024 channels
constexpr int CPB  = 128;                   // channels per block (= blockDim)
constexpr int TL   = 64;                    // L-tile per block
constexpr int HALO = 31;                    // k_max causal halo (taps look back k_max - 1)
constexpr int ROWS = TL + HALO;             // 95 staged rows
constexpr long long PLANE = (long long)Bn * Ln * Cn;  // elems per output tensor

#define AS1 __attribute__((address_space(1)))
#define AS3 __attribute__((address_space(3)))

typedef int v4i __attribute__((vector_size(16)));   // 16-byte payload type for b128

// ---------------- CDNA5 async global->LDS copy (ASYNCcnt path) ----------------
__device__ __forceinline__ void async_copy_b128(const float* g, float* l) {
#if defined(__has_builtin) && __has_builtin(__builtin_amdgcn_global_load_async_to_lds_b128)
    __builtin_amdgcn_global_load_async_to_lds_b128(
        (AS1 v4i*)(uintptr_t)g,                      // global src (AS1, printed as __device__)
        (AS3 v4i*)(unsigned)(uintptr_t)l,            // LDS dst (AS3); low 32b of generic = LDS offset
        /*offset=*/0, /*cpol=*/0);
#else
    unsigned lds_addr = (unsigned)(uintptr_t)l;
    unsigned long long gaddr = (unsigned long long)(uintptr_t)g;
    asm volatile("global_load_async_to_lds_b128 %0, %1, off"
                 :: "v"(lds_addr), "v"(gaddr) : "memory");
#endif
}

__device__ __forceinline__ void wait_async0() {
#if defined(__has_builtin) && __has_builtin(__builtin_amdgcn_s_wait_asynccnt)
    __builtin_amdgcn_s_wait_asynccnt(0);
#else
    asm volatile("s_wait_asynccnt 0x0" ::: "memory");
#endif
}

// ---------------- kernel ----------------
__global__ __launch_bounds__(CPB) void fir_ms_kernel(
    const float* __restrict__ x,
    const float* __restrict__ f3g,  const float* __restrict__ f7g,
    const float* __restrict__ f15g, const float* __restrict__ f31g,
    float* __restrict__ out)
{
    __shared__ float tile[ROWS * CPB];      // 95 * 128 * 4 = 47.5 KB

    const int t  = threadIdx.x;
    int bid = blockIdx.x;
    const int lt = bid % (Ln / TL);  bid /= (Ln / TL);
    const int cb = bid % (Cn / CPB); bid /= (Cn / CPB);
    const int b  = bid;
    const int l0 = lt * TL;
    const int c0 = cb * CPB;
    const int c  = c0 + t;

    // 1) zero causal left-halo rows (only the first L-tile needs it)
    if (l0 == 0) {
        #pragma unroll
        for (int r = 0; r < HALO; ++r) tile[r * CPB + t] = 0.0f;
    }

    // 2) async-stage rows: 128 lanes x 16B = 4 rows (512 B/row) per sweep
    {
        const int colq = t & 31;     // 16-byte column within a row
        const int rsub = t >> 5;     // 0..3: row within sweep
        const float* gx = x + ((size_t)b * Ln) * Cn + c0 + colq * 4;
        #pragma unroll
        for (int s = 0; s < (ROWS + 3) / 4; ++s) {     // 24 async b128 per lane in flight
            const int r  = s * 4 + rsub;
            const int gl = l0 - HALO + r;
            if (r < ROWS && gl >= 0) {
                async_copy_b128(gx + (size_t)gl * Cn, &tile[r * CPB + colq * 4]);
            }
        }
    }
    wait_async0();
    __syncthreads();

    // 3) per-channel filter taps -> VGPRs (tiny, L2-resident, heavily reused)
    float w3[3], w7[7], w15[15], w31[31];
    #pragma unroll
    for (int j = 0; j < 3;  ++j) w3[j]  = f3g [c * 3  + j];
    #pragma unroll
    for (int j = 0; j < 7;  ++j) w7[j]  = f7g [c * 7  + j];
    #pragma unroll
    for (int j = 0; j < 15; ++j) w15[j] = f15g[c * 15 + j];
    #pragma unroll
    for (int j = 0; j < 31; ++j) w31[j] = f31g[c * 31 + j];

    // 4) sliding register window; win[m] = x[l0 + i - 31 + m]
    float win[HALO + 8];
    #pragma unroll
    for (int m = 0; m < HALO; ++m) win[m] = tile[m * CPB + t];

    const size_t obase = ((size_t)b * Ln + l0) * Cn + c;
    for (int i = 0; i < TL; i += 8) {
        #pragma unroll
        for (int u = 0; u < 8; ++u)
            win[HALO + u] = tile[(HALO + i + u) * CPB + t];   // conflict-free ds_load

        #pragma unroll
        for (int u = 0; u < 8; ++u) {
            // output l = l0+i+u uses win[31 + u + j - (k-1)]
            float y31 = 0.f, y15 = 0.f, y7 = 0.f, y3 = 0.f;
            #pragma unroll
            for (int j = 0; j < 31; ++j) y31 = __builtin_fmaf(w31[j], win[u + 1  + j], y31);
            #pragma unroll
            for (int j = 0; j < 15; ++j) y15 = __builtin_fmaf(w15[j], win[u + 17 + j], y15);
            #pragma unroll
            for (int j = 0; j < 7;  ++j) y7  = __builtin_fmaf(w7[j],  win[u + 25 + j], y7);
            #pragma unroll
            for (int j = 0; j < 3;  ++j) y3  = __builtin_fmaf(w3[j],  win[u + 29 + j], y3);

            const size_t o = obase + (size_t)(i + u) * Cn;
            // write-once 128 MiB output stream -> nontemporal
            __builtin_nontemporal_store(y3,  out + o);
            __builtin_nontemporal_store(y7,  out + (size_t)PLANE     + o);
            __builtin_nontemporal_store(y15, out + (size_t)PLANE * 2 + o);
            __builtin_nontemporal_store(y31, out + (size_t)PLANE * 3 + o);
        }

        #pragma unroll
        for (int m = 0; m < HALO; ++m) win[m] = win[m + 8];
    }
}

// ---------------- launch ----------------
extern "C" void kernel_launch(void* const* d_in, const int* in_sizes, int n_in,
                              void* d_out, int out_size, void* d_ws, size_t ws_size,
                              hipStream_t stream) {
    const float* x    = (const float*)d_in[0];
    const float* f3   = (const float*)d_in[1];
    const float* f7   = (const float*)d_in[2];
    const float* f15  = (const float*)d_in[3];
    const float* f31  = (const float*)d_in[4];
    float* out = (float*)d_out;

    const int nblocks = Bn * (Cn / CPB) * (Ln / TL);   // 4 * 8 * 32 = 1024
    fir_ms_kernel<<<nblocks, CPB, 0, stream>>>(x, f3, f7, f15, f31, out);
    (void)in_sizes; (void)n_in; (void)out_size; (void)d_ws; (void)ws_size;
}